// SKA_54769422958614
// MI455X (gfx1250) — compile-verified
//
#include <hip/hip_runtime.h>

typedef float v4f __attribute__((ext_vector_type(4)));
typedef int   v4i __attribute__((ext_vector_type(4)));

#define GLOBAL_AS __attribute__((address_space(1)))
#define LDS_AS    __attribute__((address_space(3)))

#define B_ 8
#define C_ 256
#define H_ 64
#define W_ 64
#define G_ 32
#define CPG 8
#define KK 9
#define TR 16                       // output rows per block tile
#define LROWS (TR + 2)              // 18 staged rows (with halo)
#define LROWSTRIDE 72               // floats per LDS row: [0..2]=pad, 3=left halo,
                                    // 4..67 = image cols 0..63, 68 = right halo
#define LCH (LROWS * LROWSTRIDE)    // 1296 floats per channel
#define LDS_FLOATS (CPG * LCH)      // 10368 floats = 41472 B

#if defined(__HIP_DEVICE_COMPILE__) && __has_builtin(__builtin_amdgcn_global_load_async_to_lds_b128)
#define USE_ASYNC 1
#else
#define USE_ASYNC 0
#endif

__global__ __launch_bounds__(256) void SKA_ska_kernel(const float* __restrict__ x,
                                                      const float* __restrict__ w,
                                                      float* __restrict__ out) {
    __shared__ __align__(16) float lds[LDS_FLOATS];

    const int blk  = blockIdx.x;
    const int tile = blk & 3;          // H/TR = 4 tiles
    const int g    = (blk >> 2) & 31;
    const int b    = blk >> 7;
    const int r0   = tile * TR;
    const int tid  = threadIdx.x;

    const float* wb = w + ((b * G_ + g) * KK) * (H_ * W_);

    // ---- warm the weight tile into cache while x is staged (global_prefetch_b8)
    // 9 planes x 16 rows x 64 cols x 4B = 36 KB = 288 x 128B lines
    for (int i = tid; i < KK * TR * 2; i += 256) {
        int k    = i >> 5;             // 0..8
        int rem  = i & 31;
        int row  = rem >> 1;           // 0..15
        int half = rem & 1;            // 0..1 (two 128B halves per 256B row)
        __builtin_prefetch(wb + (k * H_ + r0 + row) * W_ + half * 32, 0, 1);
    }

    // ---- zero the staging tile: halo columns / out-of-image rows must be 0 ----
    {
        v4f z = {0.f, 0.f, 0.f, 0.f};
        for (int i = tid; i < LDS_FLOATS / 4; i += 256)
            ((v4f*)lds)[i] = z;
    }
    __syncthreads();

    // ---- stage x tile: 8 channels x 18 rows x 64 cols, async B128 per lane ----
    const int xbase = (b * C_ + g * CPG) * (H_ * W_);
    for (int i = tid; i < CPG * LROWS * 16; i += 256) {
        int ch   = i / (LROWS * 16);
        int rem  = i - ch * (LROWS * 16);
        int lrow = rem >> 4;
        int colv = rem & 15;
        int grow = r0 - 1 + lrow;
        if (grow >= 0 && grow < H_) {
            int goff = xbase + (ch * H_ + grow) * W_ + colv * 4;
            int loff = ch * LCH + lrow * LROWSTRIDE + 4 + colv * 4;  // 16B aligned
#if USE_ASYNC
            __builtin_amdgcn_global_load_async_to_lds_b128(
                (GLOBAL_AS v4i*)(const_cast<float*>(x) + goff),
                (LDS_AS v4i*)&lds[loff],
                0, 0);
#else
            *(v4f*)&lds[loff] = *(const v4f*)(x + goff);
#endif
        }
    }
#if USE_ASYNC
#if __has_builtin(__builtin_amdgcn_s_wait_asynccnt)
    __builtin_amdgcn_s_wait_asynccnt(0);
#else
    asm volatile("s_wait_asynccnt 0" ::: "memory");
#endif
#endif
    __syncthreads();

    // ---- compute: thread (col, rq) owns 4 CONSECUTIVE rows rq*4..rq*4+3 ----
    const int col   = tid & 63;
    const int rq    = tid >> 6;         // 0..3
    const int lbase = rq * 4;           // first tile-local output row
    float* ob = out + (b * C_ + g * CPG) * (H_ * W_);

    // per-pixel weights for this thread's 4 output rows (36 regs, read w once)
    float wt[4][KK];
#pragma unroll
    for (int rp = 0; rp < 4; ++rp) {
        const int pix = (r0 + lbase + rp) * W_ + col;
#pragma unroll
        for (int k = 0; k < KK; ++k)
            wt[rp][k] = wb[k * (H_ * W_) + pix];
    }

#pragma unroll
    for (int cc = 0; cc < CPG; ++cc) {
        // 6-row x 3-col register window feeds all 4 outputs (18 ds loads vs 36)
        const float* lp = lds + cc * LCH + lbase * LROWSTRIDE + (col + 3);
        float xr[6][3];
#pragma unroll
        for (int i = 0; i < 6; ++i)
#pragma unroll
            for (int j = 0; j < 3; ++j)
                xr[i][j] = lp[i * LROWSTRIDE + j];

#pragma unroll
        for (int rp = 0; rp < 4; ++rp) {
            float acc = 0.f;
#pragma unroll
            for (int i = 0; i < 3; ++i)
#pragma unroll
                for (int j = 0; j < 3; ++j)
                    acc = fmaf(xr[rp + i][j], wt[rp][i * 3 + j], acc);
            ob[cc * (H_ * W_) + (r0 + lbase + rp) * W_ + col] = acc;
        }
    }
}

extern "C" void kernel_launch(void* const* d_in, const int* in_sizes, int n_in,
                              void* d_out, int out_size, void* d_ws, size_t ws_size,
                              hipStream_t stream) {
    const float* x = (const float*)d_in[0];
    const float* w = (const float*)d_in[1];
    float* out = (float*)d_out;
    (void)in_sizes; (void)n_in; (void)out_size; (void)d_ws; (void)ws_size;

    dim3 grid(B_ * G_ * (H_ / TR));   // 8 * 32 * 4 = 1024 blocks
    dim3 block(256);                  // 8 wave32s
    SKA_ska_kernel<<<grid, block, 0, stream>>>(x, w, out);
}